// DGCRN_52707838656911
// MI455X (gfx1250) — compile-verified
//
#include <hip/hip_runtime.h>
#include <cstddef>

// ---------------- static problem config ----------------
constexpr int kB = 16, kT = 12, kQ = 12, kN = 325, kC = 2, kHID = 64;
constexpr int kGIN = 66;            // HID + C
constexpr int kGCAT = 198;          // (DEPTH+1)*GIN
constexpr int kND = 40, kHD = 16, kMD = 2;
constexpr int kNT = 21;             // ceil(325/16) row tiles
constexpr int kNP = 328;            // padded adjacency pitch (multiple of 4, zero pad cols)
constexpr int kKP = 200;            // padded K pitch for transposed RNN weights
constexpr float kALPHA = 0.05f, kBETA = 0.95f, kGAMMA = 0.95f, kTANHA = 3.0f;

typedef __attribute__((ext_vector_type(2))) float v2f;
typedef __attribute__((ext_vector_type(8))) float v8f;

// D = A(16x4,f32) * B(4x16,f32) + C  on the CDNA5 matrix pipe
__device__ __forceinline__ v8f wmma4(v2f a, v2f b, v8f c) {
    return __builtin_amdgcn_wmma_f32_16x16x4_f32(
        /*neg_a=*/false, a, /*neg_b=*/false, b,
        /*c_mod=*/(short)0, c, /*reuse_a=*/false, /*reuse_b=*/false);
}

__device__ __forceinline__ float sigf(float x) { return 1.f / (1.f + __expf(-x)); }

// ---------------- elementwise helpers ----------------
__global__ void zero_kernel(float* __restrict__ p, int n) {
    int i = blockIdx.x * blockDim.x + threadIdx.x;
    if (i < n) p[i] = 0.f;
}

// S[w*kNP + v] = gamma * adjs[v*kN + w]  (transposed, scaled, zero-padded)
__global__ void static_comb_kernel(const float* __restrict__ adjs, float* __restrict__ S,
                                   float gamma) {
    int idx = blockIdx.x * blockDim.x + threadIdx.x;
    if (idx >= kN * kNP) return;
    int v = idx % kNP, w = idx / kNP;
    S[idx] = (v < kN) ? gamma * adjs[(size_t)v * kN + w] : 0.f;
}

// rWt[g][n][k] = rW[g][k][n], K padded to kKP with zeros
__global__ void wtrans_kernel(const float* __restrict__ rW, float* __restrict__ rWt) {
    int idx = blockIdx.x * blockDim.x + threadIdx.x;
    if (idx >= 12 * kHID * kKP) return;
    int k = idx % kKP;
    int rem = idx / kKP;
    int n = rem % kHID, g = rem / kHID;
    rWt[idx] = (k < kGCAT) ? rW[((size_t)g * kGCAT + k) * kHID + n] : 0.f;
}

__global__ void build_hyperin_enc(const float* __restrict__ inp, int t,
                                  const float* __restrict__ H, float* __restrict__ hin) {
    int idx = blockIdx.x * blockDim.x + threadIdx.x;
    if (idx >= kB * kN * kGIN) return;
    int c = idx % kGIN, bn = idx / kGIN;
    float v;
    if (c < kC) {
        int n = bn % kN, b = bn / kN;
        v = inp[(((size_t)b * kT + t) * kN + n) * kC + c];
    } else {
        v = H[(size_t)bn * kHID + (c - kC)];
    }
    hin[idx] = v;
}

__global__ void build_hyperin_dec(const float* __restrict__ ycl, int q,
                                  const float* __restrict__ go,
                                  const float* __restrict__ H, float* __restrict__ hin) {
    int idx = blockIdx.x * blockDim.x + threadIdx.x;
    if (idx >= kB * kN * kGIN) return;
    int c = idx % kGIN, bn = idx / kGIN;
    float v;
    if (c == 0) {
        v = go[bn];
    } else if (c == 1) {
        int n = bn % kN, b = bn / kN;
        v = ycl[(((size_t)b * kQ + q) * kN + n) * kC + 1];
    } else {
        v = H[(size_t)bn * kHID + (c - kC)];
    }
    hin[idx] = v;
}

__global__ void build_temp_kernel(const float* __restrict__ hin, const float* __restrict__ r,
                                  const float* __restrict__ H, float* __restrict__ tmpc) {
    int idx = blockIdx.x * blockDim.x + threadIdx.x;
    if (idx >= kB * kN * kGIN) return;
    int c = idx % kGIN, bn = idx / kGIN;
    float v;
    if (c < kC) v = hin[idx];                                   // xt channels
    else v = r[(size_t)bn * kHID + (c - kC)] * H[(size_t)bn * kHID + (c - kC)];
    tmpc[idx] = v;
}

// copy hop-0 (the raw input) into channels [0,66) of a 198-wide concat buffer
__global__ void seed_kernel(const float* __restrict__ src, float* __restrict__ ho) {
    int idx = blockIdx.x * blockDim.x + threadIdx.x;
    if (idx >= kB * kN * kGIN) return;
    int c = idx % kGIN, bn = idx / kGIN;
    ho[(size_t)bn * kGCAT + c] = src[idx];
}

__global__ void hupdate_kernel(const float* __restrict__ z, const float* __restrict__ Cn,
                               float* __restrict__ H) {
    int idx = blockIdx.x * blockDim.x + threadIdx.x;
    if (idx >= kB * kN * kHID) return;
    float zz = z[idx];
    H[idx] = zz * H[idx] + (1.f - zz) * Cn[idx];
}

__global__ void final_kernel(const float* __restrict__ H, const float* __restrict__ Wf,
                             const float* __restrict__ bf, float* __restrict__ go,
                             float* __restrict__ out, int q) {
    int idx = blockIdx.x * blockDim.x + threadIdx.x;
    if (idx >= kB * kN) return;
    const float* h = H + (size_t)idx * kHID;
    float s = bf[0];
    for (int c = 0; c < kHID; ++c) s += h[c] * Wf[c];
    go[idx] = s;
    int b = idx / kN, n = idx % kN;
    out[((size_t)b * kQ + q) * kN + n] = s;     // (B, Q, N)
}

// ---------------- WMMA kernel 1: graph-hop propagation ----------------
// ho[b, w, outOff+c] = ALPHA * x0[b,w,c] + sum_v ho[b, v, inOff+c] * comb[w, v]
// comb is the pre-combined transposed adjacency (pitch kNP, zero pad cols 325..327,
// which makes the K tail exact with no guards: stray B rows multiply by 0).
template <bool BATCHED>
__global__ void __launch_bounds__(32)
hop_kernel(const float* __restrict__ x0, float* __restrict__ ho,
           int inOff, int outOff, const float* __restrict__ comb) {
    const int b = blockIdx.x;
    const int w0 = blockIdx.y * 16;
    const int lane = threadIdx.x;
    const int half = lane >> 4;
    const int mn = lane & 15;

    const float* cb = BATCHED ? comb + (size_t)b * kN * kNP : comb;
    const int wc = min(w0 + mn, kN - 1);                    // clamp: bad rows never stored
    const float* arow = cb + (size_t)wc * kNP + 2 * half;   // 8B-aligned v2f loads
    const float* hinB = ho + (size_t)b * kN * kGCAT + inOff;

    int ccol[5];
    #pragma unroll
    for (int ct = 0; ct < 5; ++ct) ccol[ct] = min(ct * 16 + mn, kGIN - 1);

    v8f acc[5] = {};
    #pragma unroll 2
    for (int k0 = 0; k0 < kN; k0 += 4) {                    // 82 fully branchless chunks
        const v2f a = *(const v2f*)(arow + k0);
        const float* h0 = hinB + (size_t)(k0 + 2 * half) * kGCAT;
        #pragma unroll
        for (int ct = 0; ct < 5; ++ct) {
            v2f bv;
            bv.x = h0[ccol[ct]];
            bv.y = h0[kGCAT + ccol[ct]];
            acc[ct] = wmma4(a, bv, acc[ct]);
        }
    }

    float* outB = ho + (size_t)b * kN * kGCAT + outOff;
    const float* x0B = x0 + (size_t)b * kN * kGIN;
    if (w0 + 16 <= kN) {                                    // fast path: full row tile
        #pragma unroll
        for (int ct = 0; ct < 4; ++ct) {                    // c = 0..63 always valid
            const int c = ct * 16 + mn;
            #pragma unroll
            for (int rr = 0; rr < 8; ++rr) {
                const int w = w0 + rr + 8 * half;
                outB[(size_t)w * kGCAT + c] = kALPHA * x0B[(size_t)w * kGIN + c] + acc[ct][rr];
            }
        }
        if (mn < 2) {                                       // ct=4: c = 64..65 only
            const int c = 64 + mn;
            #pragma unroll
            for (int rr = 0; rr < 8; ++rr) {
                const int w = w0 + rr + 8 * half;
                outB[(size_t)w * kGCAT + c] = kALPHA * x0B[(size_t)w * kGIN + c] + acc[4][rr];
            }
        }
    } else {                                                // edge tile: guarded
        #pragma unroll
        for (int ct = 0; ct < 5; ++ct) {
            const int c = ct * 16 + mn;
            if (c >= kGIN) continue;
            #pragma unroll
            for (int rr = 0; rr < 8; ++rr) {
                const int w = w0 + rr + 8 * half;
                if (w < kN)
                    outB[(size_t)w * kGCAT + c] = kALPHA * x0B[(size_t)w * kGIN + c] + acc[ct][rr];
            }
        }
    }
}

// ---------------- WMMA kernel 2: dynamic graph nv1 @ nv2^T ----------------
__global__ void __launch_bounds__(32)
dyngraph_kernel(const float* __restrict__ nv1, const float* __restrict__ nv2,
                float* __restrict__ Araw) {
    const int b = blockIdx.x;
    const int i0 = blockIdx.y * 16, j0 = blockIdx.z * 16;
    const int lane = threadIdx.x;
    const int half = lane >> 4;
    const int mn = lane & 15;

    const int ic = min(i0 + mn, kN - 1);
    const int jc = min(j0 + mn, kN - 1);
    const float* arow = nv1 + (size_t)b * kN * kND + (size_t)ic * kND + 2 * half;
    const float* brow = nv2 + (size_t)b * kN * kND + (size_t)jc * kND + 2 * half;

    v8f acc = {};
    #pragma unroll
    for (int k0 = 0; k0 < kND; k0 += 4) {               // K=40, fully branchless
        const v2f a = *(const v2f*)(arow + k0);
        const v2f bv = *(const v2f*)(brow + k0);
        acc = wmma4(a, bv, acc);
    }
    float* Ab = Araw + (size_t)b * kN * kN;
    const int j = j0 + mn;
    if (i0 + 16 <= kN && j0 + 16 <= kN) {               // fast path: full tile
        #pragma unroll
        for (int rr = 0; rr < 8; ++rr) {
            const int ii = i0 + rr + 8 * half;
            Ab[(size_t)ii * kN + j] = acc[rr];
        }
    } else {
        #pragma unroll
        for (int rr = 0; rr < 8; ++rr) {
            const int ii = i0 + rr + 8 * half;
            if (ii < kN && j < kN) Ab[(size_t)ii * kN + j] = acc[rr];
        }
    }
}

// ---------------- row-norm + fold static adjacency into transposed combined mats ----
// combA[b][w*kNP+v] = BETA*norm(adj )[v,w] + GAMMA*adj0[v,w]
// combB[b][w*kNP+v] = BETA*norm(adjT)[v,w] + GAMMA*adj1[v,w]
__global__ void __launch_bounds__(256)
adjnorm_comb_kernel(const float* __restrict__ Araw,
                    const float* __restrict__ adj0, const float* __restrict__ adj1,
                    float* __restrict__ combA, float* __restrict__ combB) {
    __shared__ float tv[kN];
    __shared__ float redp[256], redn[256];
    const int bv = blockIdx.x;
    const int b = bv / kN, v = bv % kN;
    const int tid = threadIdx.x;
    const float* Ab = Araw + (size_t)b * kN * kN;

    float sp = 0.f, sn = 0.f;
    for (int w = tid; w < kN; w += 256) {
        float t = tanhf(kTANHA * (Ab[(size_t)v * kN + w] - Ab[(size_t)w * kN + v]));
        tv[w] = t;
        sp += fmaxf(t, 0.f);
        sn += fmaxf(-t, 0.f);
    }
    redp[tid] = sp; redn[tid] = sn;
    __syncthreads();
    for (int s = 128; s > 0; s >>= 1) {
        if (tid < s) { redp[tid] += redp[tid + s]; redn[tid] += redn[tid + s]; }
        __syncthreads();
    }
    const float isp = 1.f / (1.f + redp[0]);
    const float isn = 1.f / (1.f + redn[0]);
    float* cA = combA + (size_t)b * kN * kNP;
    float* cB = combB + (size_t)b * kN * kNP;
    for (int w = tid; w < kN; w += 256) {
        float t = tv[w];
        float eye = (w == v) ? 1.f : 0.f;
        float ap = (fmaxf(t, 0.f) + eye) * isp;    // norm(adj)[v,w]
        float at = (fmaxf(-t, 0.f) + eye) * isn;   // norm(adjT)[v,w]
        cA[(size_t)w * kNP + v] = kBETA * ap + kGAMMA * adj0[(size_t)v * kN + w];
        cB[(size_t)w * kNP + v] = kBETA * at + kGAMMA * adj1[(size_t)v * kN + w];
    }
}

// ---------------- hyper MLPs (4x 198->16->2->40) + node vectors ----------------
__global__ void __launch_bounds__(128)
hypermlp_kernel(const float* __restrict__ ho0, const float* __restrict__ ho1,
                const float* __restrict__ hW1, const float* __restrict__ hb1,
                const float* __restrict__ hW2, const float* __restrict__ hb2,
                const float* __restrict__ hW3, const float* __restrict__ hb3,
                const float* __restrict__ emb1, int gbase,
                float* __restrict__ nv1, float* __restrict__ nv2) {
    __shared__ float xr[2][kGCAT];
    __shared__ float h1[4][kHD];
    __shared__ float h2[4][kMD];
    const int bn = blockIdx.x;
    const int tid = threadIdx.x;
    const int node = bn % kN;

    const float* r0 = ho0 + (size_t)bn * kGCAT;
    const float* r1 = ho1 + (size_t)bn * kGCAT;
    for (int c = tid; c < kGCAT; c += 128) { xr[0][c] = r0[c]; xr[1][c] = r1[c]; }
    __syncthreads();

    if (tid < 64) {                      // 4 MLPs x 16 hidden units
        const int g = tid >> 4, u = tid & 15;
        const int gg = gbase + g;
        const float* W = hW1 + (size_t)gg * kGCAT * kHD;
        const float* xx = xr[g & 1];     // g0,g2 read adj0-concat; g1,g3 read adj1-concat
        float s = hb1[gg * kHD + u];
        for (int c = 0; c < kGCAT; ++c) s += xx[c] * W[(size_t)c * kHD + u];
        h1[g][u] = sigf(s);
    }
    __syncthreads();
    if (tid < 8) {                       // 4 MLPs x 2 mid units
        const int g = tid >> 1, u = tid & 1;
        const int gg = gbase + g;
        const float* W = hW2 + (size_t)gg * kHD * kMD;
        float s = hb2[gg * kMD + u];
        for (int c = 0; c < kHD; ++c) s += h1[g][c] * W[c * kMD + u];
        h2[g][u] = sigf(s);
    }
    __syncthreads();
    if (tid < 2 * kND) {                 // f1 -> nv1, f2 -> nv2
        const int f = tid / kND, d = tid % kND;
        const int gA = gbase + 2 * f, gB = gbase + 2 * f + 1;
        const float* WA = hW3 + (size_t)gA * kMD * kND;
        const float* WB = hW3 + (size_t)gB * kMD * kND;
        float s = hb3[gA * kND + d] + hb3[gB * kND + d];
        for (int m = 0; m < kMD; ++m)
            s += h2[2 * f][m] * WA[m * kND + d] + h2[2 * f + 1][m] * WB[m * kND + d];
        float val = tanhf(kTANHA * emb1[node * kND + d] * s);
        (f == 0 ? nv1 : nv2)[(size_t)bn * kND + d] = val;
    }
}

// ---------------- WMMA kernel 3: z,r = sigmoid(hoA@W + hoB@W' + b) ----------------
// rWt is the transposed, K-padded weight bank: fully branchless 50-chunk loop,
// all operands loaded as aligned v2f.
__global__ void __launch_bounds__(32)
gates_kernel(const float* __restrict__ hoA, const float* __restrict__ hoB,
             const float* __restrict__ rWt, const float* __restrict__ rb,
             int g0, float* __restrict__ z, float* __restrict__ r) {
    const int lane = threadIdx.x;
    const int half = lane >> 4;
    const int mn = lane & 15;
    const int row0 = blockIdx.x * 16;      // B*N = 5200 = 325 exact tiles
    const int m = row0 + mn;

    const float* aArow = hoA + (size_t)m * kGCAT + 2 * half;
    const float* aBrow = hoB + (size_t)m * kGCAT + 2 * half;
    const float* w0r[4]; const float* w1r[4]; const float* w2r[4]; const float* w3r[4];
    #pragma unroll
    for (int ct = 0; ct < 4; ++ct) {
        const int n = ct * 16 + mn;
        w0r[ct] = rWt + ((size_t)(g0 + 0) * kHID + n) * kKP + 2 * half;
        w1r[ct] = rWt + ((size_t)(g0 + 1) * kHID + n) * kKP + 2 * half;
        w2r[ct] = rWt + ((size_t)(g0 + 2) * kHID + n) * kKP + 2 * half;
        w3r[ct] = rWt + ((size_t)(g0 + 3) * kHID + n) * kKP + 2 * half;
    }

    v8f accz[4] = {}; v8f accr[4] = {};
    for (int k0 = 0; k0 < kGCAT; k0 += 4) {    // 50 branchless chunks (pad weights = 0)
        const v2f aA = *(const v2f*)(aArow + k0);
        const v2f aB = *(const v2f*)(aBrow + k0);
        #pragma unroll
        for (int ct = 0; ct < 4; ++ct) {
            accz[ct] = wmma4(aA, *(const v2f*)(w0r[ct] + k0), accz[ct]);
            accz[ct] = wmma4(aB, *(const v2f*)(w1r[ct] + k0), accz[ct]);
            accr[ct] = wmma4(aA, *(const v2f*)(w2r[ct] + k0), accr[ct]);
            accr[ct] = wmma4(aB, *(const v2f*)(w3r[ct] + k0), accr[ct]);
        }
    }
    #pragma unroll
    for (int ct = 0; ct < 4; ++ct) {
        const int n = ct * 16 + mn;
        const float bz = rb[(g0 + 0) * kHID + n] + rb[(g0 + 1) * kHID + n];
        const float br = rb[(g0 + 2) * kHID + n] + rb[(g0 + 3) * kHID + n];
        #pragma unroll
        for (int rr = 0; rr < 8; ++rr) {
            const int row = row0 + rr + 8 * half;
            z[(size_t)row * kHID + n] = sigf(accz[ct][rr] + bz);
            r[(size_t)row * kHID + n] = sigf(accr[ct][rr] + br);
        }
    }
}

// ---------------- WMMA kernel 4: Cn = tanh(hoC@W + hoD@W' + b) ----------------
__global__ void __launch_bounds__(32)
cn_kernel(const float* __restrict__ hoC, const float* __restrict__ hoD,
          const float* __restrict__ rWt, const float* __restrict__ rb,
          int g0, float* __restrict__ Cn) {
    const int lane = threadIdx.x;
    const int half = lane >> 4;
    const int mn = lane & 15;
    const int row0 = blockIdx.x * 16;
    const int m = row0 + mn;

    const float* aArow = hoC + (size_t)m * kGCAT + 2 * half;
    const float* aBrow = hoD + (size_t)m * kGCAT + 2 * half;
    const float* w0r[4]; const float* w1r[4];
    #pragma unroll
    for (int ct = 0; ct < 4; ++ct) {
        const int n = ct * 16 + mn;
        w0r[ct] = rWt + ((size_t)(g0 + 0) * kHID + n) * kKP + 2 * half;
        w1r[ct] = rWt + ((size_t)(g0 + 1) * kHID + n) * kKP + 2 * half;
    }

    v8f acc[4] = {};
    #pragma unroll 2
    for (int k0 = 0; k0 < kGCAT; k0 += 4) {    // 50 branchless chunks
        const v2f aA = *(const v2f*)(aArow + k0);
        const v2f aB = *(const v2f*)(aBrow + k0);
        #pragma unroll
        for (int ct = 0; ct < 4; ++ct) {
            acc[ct] = wmma4(aA, *(const v2f*)(w0r[ct] + k0), acc[ct]);
            acc[ct] = wmma4(aB, *(const v2f*)(w1r[ct] + k0), acc[ct]);
        }
    }
    #pragma unroll
    for (int ct = 0; ct < 4; ++ct) {
        const int n = ct * 16 + mn;
        const float bc = rb[(g0 + 0) * kHID + n] + rb[(g0 + 1) * kHID + n];
        #pragma unroll
        for (int rr = 0; rr < 8; ++rr) {
            const int row = row0 + rr + 8 * half;
            Cn[(size_t)row * kHID + n] = tanhf(acc[ct][rr] + bc);
        }
    }
}

// ---------------- host driver ----------------
extern "C" void kernel_launch(void* const* d_in, const int* in_sizes, int n_in,
                              void* d_out, int out_size, void* d_ws, size_t ws_size,
                              hipStream_t stream) {
    (void)in_sizes; (void)n_in; (void)out_size; (void)ws_size;
    const float* inp  = (const float*)d_in[0];
    const float* ycl  = (const float*)d_in[1];
    const float* adj0 = (const float*)d_in[2];
    const float* adj1 = (const float*)d_in[3];
    const float* emb1 = (const float*)d_in[4];
    const float* hW1  = (const float*)d_in[5];
    const float* hb1  = (const float*)d_in[6];
    const float* hW2  = (const float*)d_in[7];
    const float* hb2  = (const float*)d_in[8];
    const float* hW3  = (const float*)d_in[9];
    const float* hb3  = (const float*)d_in[10];
    const float* rW   = (const float*)d_in[11];
    const float* rb   = (const float*)d_in[12];
    const float* Wf   = (const float*)d_in[13];
    const float* bf   = (const float*)d_in[14];
    float* out = (float*)d_out;

    float* ws = (float*)d_ws;
    size_t off = 0;
    auto alloc = [&](size_t n) { float* p = ws + off; off += n; return p; };
    float* H     = alloc((size_t)kB * kN * kHID);
    float* hin   = alloc((size_t)kB * kN * kGIN);
    float* tmpc  = alloc((size_t)kB * kN * kGIN);
    float* ho0   = alloc((size_t)kB * kN * kGCAT);
    float* ho1   = alloc((size_t)kB * kN * kGCAT);
    float* ho2   = alloc((size_t)kB * kN * kGCAT);
    float* ho3   = alloc((size_t)kB * kN * kGCAT);
    float* nv1   = alloc((size_t)kB * kN * kND);
    float* nv2   = alloc((size_t)kB * kN * kND);
    float* Araw  = alloc((size_t)kB * kN * kN);
    float* S0    = alloc((size_t)kN * kNP);            // gamma*adj0^T (padded)
    float* S1    = alloc((size_t)kN * kNP);            // gamma*adj1^T (padded)
    float* combA = alloc((size_t)kB * kN * kNP);       // beta*adp^T + gamma*adj0^T
    float* combB = alloc((size_t)kB * kN * kNP);       // beta*adpT^T + gamma*adj1^T
    float* rWt   = alloc((size_t)12 * kHID * kKP);     // transposed, K-padded RNN weights
    float* zB    = alloc((size_t)kB * kN * kHID);
    float* rB    = alloc((size_t)kB * kN * kHID);
    float* CnB   = alloc((size_t)kB * kN * kHID);
    float* go    = alloc((size_t)kB * kN);

    auto ew = [](int n) { return dim3((unsigned)((n + 255) / 256)); };

    // one-time init: H=0, go=0, comb pads=0 (pad cols never rewritten), static combos,
    // transposed weight bank
    zero_kernel<<<ew(kB * kN * kHID), 256, 0, stream>>>(H, kB * kN * kHID);
    zero_kernel<<<ew(kB * kN), 256, 0, stream>>>(go, kB * kN);
    zero_kernel<<<ew(kB * kN * kNP), 256, 0, stream>>>(combA, kB * kN * kNP);
    zero_kernel<<<ew(kB * kN * kNP), 256, 0, stream>>>(combB, kB * kN * kNP);
    static_comb_kernel<<<ew(kN * kNP), 256, 0, stream>>>(adj0, S0, kGAMMA);
    static_comb_kernel<<<ew(kN * kNP), 256, 0, stream>>>(adj1, S1, kGAMMA);
    wtrans_kernel<<<ew(12 * kHID * kKP), 256, 0, stream>>>(rW, rWt);

    const dim3 hopg(kB, kNT);
    const dim3 dyng(kB, kNT, kNT);
    const int nEW = kB * kN * kGIN;

    auto step = [&](bool dec, int tq) {
        if (!dec) build_hyperin_enc<<<ew(nEW), 256, 0, stream>>>(inp, tq, H, hin);
        else      build_hyperin_dec<<<ew(nEW), 256, 0, stream>>>(ycl, tq, go, H, hin);

        // hyper hop-concats (static adjacency only)
        seed_kernel<<<ew(nEW), 256, 0, stream>>>(hin, ho0);
        seed_kernel<<<ew(nEW), 256, 0, stream>>>(hin, ho1);
        hop_kernel<false><<<hopg, 32, 0, stream>>>(hin, ho0, 0, kGIN, S0);
        hop_kernel<false><<<hopg, 32, 0, stream>>>(hin, ho0, kGIN, 2 * kGIN, S0);
        hop_kernel<false><<<hopg, 32, 0, stream>>>(hin, ho1, 0, kGIN, S1);
        hop_kernel<false><<<hopg, 32, 0, stream>>>(hin, ho1, kGIN, 2 * kGIN, S1);

        hypermlp_kernel<<<kB * kN, 128, 0, stream>>>(ho0, ho1, hW1, hb1, hW2, hb2,
                                                     hW3, hb3, emb1, dec ? 4 : 0, nv1, nv2);

        dyngraph_kernel<<<dyng, 32, 0, stream>>>(nv1, nv2, Araw);
        adjnorm_comb_kernel<<<kB * kN, 256, 0, stream>>>(Araw, adj0, adj1, combA, combB);

        // RNN hop-concats for z/r; ho0/ho1 hop-0 already holds hyper_in
        hop_kernel<true><<<hopg, 32, 0, stream>>>(hin, ho0, 0, kGIN, combA);
        hop_kernel<true><<<hopg, 32, 0, stream>>>(hin, ho0, kGIN, 2 * kGIN, combA);
        hop_kernel<true><<<hopg, 32, 0, stream>>>(hin, ho1, 0, kGIN, combB);
        hop_kernel<true><<<hopg, 32, 0, stream>>>(hin, ho1, kGIN, 2 * kGIN, combB);

        const int ro = dec ? 6 : 0;
        gates_kernel<<<(kB * kN) / 16, 32, 0, stream>>>(ho0, ho1, rWt, rb, ro, zB, rB);

        build_temp_kernel<<<ew(nEW), 256, 0, stream>>>(hin, rB, H, tmpc);
        seed_kernel<<<ew(nEW), 256, 0, stream>>>(tmpc, ho2);
        seed_kernel<<<ew(nEW), 256, 0, stream>>>(tmpc, ho3);
        hop_kernel<true><<<hopg, 32, 0, stream>>>(tmpc, ho2, 0, kGIN, combA);
        hop_kernel<true><<<hopg, 32, 0, stream>>>(tmpc, ho2, kGIN, 2 * kGIN, combA);
        hop_kernel<true><<<hopg, 32, 0, stream>>>(tmpc, ho3, 0, kGIN, combB);
        hop_kernel<true><<<hopg, 32, 0, stream>>>(tmpc, ho3, kGIN, 2 * kGIN, combB);

        cn_kernel<<<(kB * kN) / 16, 32, 0, stream>>>(ho2, ho3, rWt, rb, ro + 4, CnB);
        hupdate_kernel<<<ew(kB * kN * kHID), 256, 0, stream>>>(zB, CnB, H);

        if (dec) final_kernel<<<ew(kB * kN), 256, 0, stream>>>(H, Wf, bf, go, out, tq);
    };

    for (int t = 0; t < kT; ++t) step(false, t);
    for (int q = 0; q < kQ; ++q) step(true, q);
}